// ProductBasisBlock_8675833938664
// MI455X (gfx1250) — compile-verified
//
#include <hip/hip_runtime.h>
#include <hip/hip_bf16.h>

// ---------------------------------------------------------------------------
// MACE ProductBasisBlock for MI455X (gfx1250, wave32, WMMA f32 16x16x4)
//
// B=256 nodes, C=128 channels, LM=16, O=16, K1=8, K2=32, K3=128
//
// Pipeline:
//   Kernel A: P_t[c][opqi] = sum_k U3[opqi,k] * w3[k,c]      (65536x128x128 GEMM)
//   Kernel B: per channel c:
//       out3[opq,b] = sum_i P_c[opq,i] * nf_c[i,b]           (WMMA, fused epilogue)
//       out2[b,op]  = sum_q (c2_c[op,q]+out3[opq,b])*nf_c[q,b]
//       out1[b,o]   = sum_p (c1_c[o,p]+out2[b,o*16+p])*nf_c[p,b]
//   Kernel C: y[b,d,io] = (1/sqrt(C)) sum_c out1[c,io,b]*Wlin[l(io)][c,d] + sc
// ---------------------------------------------------------------------------

typedef __attribute__((ext_vector_type(2))) float v2f;
typedef __attribute__((ext_vector_type(8))) float v8f;

__device__ __forceinline__ v8f wmma_f32_16x16x4(v2f a, v2f b, v8f c) {
  // D = A(16x4 f32) * B(4x16 f32) + C(16x16 f32)
  return __builtin_amdgcn_wmma_f32_16x16x4_f32(false, a, false, b, (short)0, c,
                                               false, false);
}

// ---------------------------------------------------------------------------
// Kernel A: P_t[c][m] = sum_k U3[m*128+k] * w3[k*128+c],  m = opqi in [0,65536)
// grid = 512 blocks x 256 threads (8 waves).
// Wave tile: 32(M) x 64(N) => 2x4 WMMA tiles, K-loop of 32 steps.
// w3 staged in LDS as K-pair interleaved: w3p[k/2][n][2] so a B fragment is a
// single aligned ds_load_b64 straight into a WMMA operand register pair.
// ---------------------------------------------------------------------------
__global__ __launch_bounds__(256) void pb_u3w3_gemm(const float* __restrict__ U3,
                                                    const float* __restrict__ w3,
                                                    float* __restrict__ Pt) {
  extern __shared__ float w3p[];  // [64 kp][128 n][2] = 16384 floats = 64 KB
  const int tid = threadIdx.x;
  for (int t = tid; t < 128 * 128; t += 256) {
    const int k = t >> 7;
    const int n = t & 127;
    w3p[(k >> 1) * 256 + n * 2 + (k & 1)] = w3[t];
  }
  __syncthreads();

  const int lane = tid & 31;
  const int wave = tid >> 5;
  const int ln = lane & 15;  // M row within tile (A) / N col (B,D)
  const int hi = lane >> 4;  // K-pair select (A,B) / M+8 select (D)

  const int nhalf = blockIdx.x & 1;           // N half: cols [nhalf*64, +64)
  const int mgrp = blockIdx.x >> 1;           // 256 M-groups of 256 rows
  const int m0 = (mgrp * 8 + wave) * 32;      // 32 M rows per wave
  const int nb = nhalf * 64;

  const float* Arow0 = U3 + (size_t)(m0 + ln) * 128;
  const float* Arow1 = U3 + (size_t)(m0 + 16 + ln) * 128;
  __builtin_prefetch(Arow0, 0, 0);
  __builtin_prefetch(Arow1, 0, 0);

  v8f acc[2][4];
#pragma unroll
  for (int mt = 0; mt < 2; ++mt)
#pragma unroll
    for (int nt = 0; nt < 4; ++nt) acc[mt][nt] = {};

  for (int k0 = 0; k0 < 128; k0 += 4) {
    const int kk = k0 + hi * 2;
    const v2f a0 = *(const v2f*)(Arow0 + kk);
    const v2f a1 = *(const v2f*)(Arow1 + kk);
    // B fragment: lane ln = col n, pair (B[kk][n], B[kk+1][n]) contiguous.
    const float* bbase = w3p + (size_t)(k0 / 2 + hi) * 256 + (nb + ln) * 2;
#pragma unroll
    for (int nt = 0; nt < 4; ++nt) {
      const v2f bf = *(const v2f*)(bbase + nt * 32);  // 16 cols * 2
      acc[0][nt] = wmma_f32_16x16x4(a0, bf, acc[0][nt]);
      acc[1][nt] = wmma_f32_16x16x4(a1, bf, acc[1][nt]);
    }
  }

#pragma unroll
  for (int mt = 0; mt < 2; ++mt) {
#pragma unroll
    for (int nt = 0; nt < 4; ++nt) {
      const int ccol = nb + nt * 16 + ln;
      float* dst = Pt + (size_t)ccol * 65536 + m0 + mt * 16 + hi * 8;
      const v8f v = acc[mt][nt];
      *(float4*)(dst) = make_float4(v[0], v[1], v[2], v[3]);
      *(float4*)(dst + 4) = make_float4(v[4], v[5], v[6], v[7]);
    }
  }
}

// ---------------------------------------------------------------------------
// Kernel B: per-channel fused corr3 -> corr2 -> corr1.
// grid = 128 blocks (one per channel) x 256 threads (8 waves).
// Dynamic LDS: out2[256b][256op] | nf[256b][16i] | c2[256op][16i] | c1[16o][16i]
// ---------------------------------------------------------------------------
__global__ __launch_bounds__(256) void pb_channel_fused(
    const float* __restrict__ nf,   // [B][C][16]
    const float* __restrict__ U1,   // [16][16][8]
    const float* __restrict__ U2,   // [16][16][16][32]
    const float* __restrict__ w1,   // [8][128]
    const float* __restrict__ w2,   // [32][128]
    const float* __restrict__ Pt,   // [C][65536]
    float* __restrict__ out1_ws) {  // [C][16 o][256 b]
  extern __shared__ float smem[];
  float* out2_lds = smem;            // 65536 floats
  float* nf_lds = out2_lds + 65536;  // 4096 floats
  float* c2_lds = nf_lds + 4096;     // 4096 floats
  float* c1_lds = c2_lds + 4096;     // 256 floats

  const int c = blockIdx.x;
  const int tid = threadIdx.x;

  // ---- Phase 0: stage nf_c, compute c2_c = U2.w2[:,c], c1_c = U1.w1[:,c] ----
  for (int t = tid; t < 4096; t += 256)
    nf_lds[t] = nf[(size_t)(t >> 4) * 2048 + c * 16 + (t & 15)];

  for (int e = tid; e < 4096; e += 256) {  // e = (o*16+p)*16 + i
    float s = 0.f;
    const float* u = U2 + (size_t)e * 32;
#pragma unroll
    for (int k = 0; k < 32; ++k) s += u[k] * w2[k * 128 + c];
    c2_lds[e] = s;
  }
  if (tid < 256) {  // e = o*16 + i
    float s = 0.f;
    const float* u = U1 + (size_t)tid * 8;
#pragma unroll
    for (int k = 0; k < 8; ++k) s += u[k] * w1[k * 128 + c];
    c1_lds[tid] = s;
  }
  __syncthreads();

  // ---- Phase 1: corr3 WMMA GEMM + fused corr2 epilogue ----
  const int lane = tid & 31;
  const int wave = tid >> 5;
  const int ln = lane & 15;
  const int hi = lane >> 4;
  const float* Pc = Pt + (size_t)c * 65536;

  for (int op = wave; op < 256; op += 8) {
    // A fragments for this op tile: rows q=0..15, K=i (16), 4 k-steps.
    v2f a[4];
    const int abase = op * 256 + ln * 16;  // (op*16 + q) * 16
#pragma unroll
    for (int ks = 0; ks < 4; ++ks)
      a[ks] = *(const v2f*)(Pc + abase + ks * 4 + hi * 2);

    const float* c2row = c2_lds + op * 16 + hi * 8;

    for (int bt = 0; bt < 16; ++bt) {
      const int b0 = bt * 16;
      v8f acc = {};
#pragma unroll
      for (int ks = 0; ks < 4; ++ks) {
        const v2f bf =
            *(const v2f*)(nf_lds + (b0 + ln) * 16 + ks * 4 + hi * 2);
        acc = wmma_f32_16x16x4(a[ks], bf, acc);
      }
      // Epilogue: out2[b,op] = sum_q (c2[op,q] + out3[q,b]) * nf[b,q].
      // Lane holds q = hi*8 + j for b = b0 + ln; pair with lane^16 for full q.
      const float* nfrow = nf_lds + (b0 + ln) * 16 + hi * 8;
      float partial = 0.f;
#pragma unroll
      for (int j = 0; j < 8; ++j) partial += (c2row[j] + acc[j]) * nfrow[j];
      const float tot = partial + __shfl_xor(partial, 16, 32);
      if (hi == 0) out2_lds[(b0 + ln) * 256 + op] = tot;
    }
  }
  __syncthreads();

  // ---- Phase 2: corr1. One thread per node b. ----
  {
    const int b = tid;
    const float* nfrow = nf_lds + b * 16;
#pragma unroll
    for (int o = 0; o < 16; ++o) {
      const float* c1row = c1_lds + o * 16;
      const float* o2row = out2_lds + b * 256 + o * 16;
      float s = 0.f;
#pragma unroll
      for (int p = 0; p < 16; ++p) s += (c1row[p] + o2row[p]) * nfrow[p];
      out1_ws[(size_t)c * 4096 + o * 256 + b] = s;  // coalesced in b
    }
  }
}

// ---------------------------------------------------------------------------
// Kernel C: o3_linear + sc.  Rows m = io*256 + b (io-major => fixed l per tile).
// grid = 256 blocks (M tiles) x 256 threads (8 waves = 8 N tiles of 16).
// ---------------------------------------------------------------------------
__global__ __launch_bounds__(256) void pb_o3_linear(
    const float* __restrict__ out1_ws,  // [C][16 io][256 b]  (A: [k][m])
    const float* __restrict__ Wlin,     // [4][128][128]
    const float* __restrict__ sc,       // [B][C][16]
    float* __restrict__ out) {          // [B][C][16]
  const int lane = threadIdx.x & 31;
  const int wave = threadIdx.x >> 5;
  const int ln = lane & 15;
  const int hi = lane >> 4;

  const int m0 = blockIdx.x * 16;
  const int io = m0 >> 8;  // fixed within tile (256 % 16 == 0)
  const int b0 = m0 & 255;
  const int l = (io == 0) ? 0 : (io < 4) ? 1 : (io < 9) ? 2 : 3;
  const float* WL = Wlin + (size_t)l * 16384;

  const int m = m0 + ln;
  const int n0 = wave * 16;
  v8f acc = {};
  for (int k0 = 0; k0 < 128; k0 += 4) {
    const int kk = k0 + hi * 2;
    v2f a;
    a.x = out1_ws[(size_t)kk * 4096 + m];
    a.y = out1_ws[(size_t)(kk + 1) * 4096 + m];
    v2f bf;
    bf.x = WL[kk * 128 + n0 + ln];
    bf.y = WL[(kk + 1) * 128 + n0 + ln];
    acc = wmma_f32_16x16x4(a, bf, acc);
  }

  const float scale = 0.08838834764831845f;  // 1/sqrt(128)
  const int d = n0 + ln;
#pragma unroll
  for (int j = 0; j < 8; ++j) {
    const int b = b0 + hi * 8 + j;
    const size_t idx = (size_t)b * 2048 + d * 16 + io;
    out[idx] = acc[j] * scale + sc[idx];
  }
}

// ---------------------------------------------------------------------------
extern "C" void kernel_launch(void* const* d_in, const int* in_sizes, int n_in,
                              void* d_out, int out_size, void* d_ws,
                              size_t ws_size, hipStream_t stream) {
  (void)in_sizes; (void)n_in; (void)out_size; (void)ws_size;
  const float* nf   = (const float*)d_in[0];  // [256][128][16]
  const float* sc   = (const float*)d_in[1];  // [256][128][16]
  const float* U1   = (const float*)d_in[2];  // [16][16][8]
  const float* U2   = (const float*)d_in[3];  // [16][16][16][32]
  const float* U3   = (const float*)d_in[4];  // [16][16][16][16][128]
  const float* w1   = (const float*)d_in[5];  // [8][128]
  const float* w2   = (const float*)d_in[6];  // [32][128]
  const float* w3   = (const float*)d_in[7];  // [128][128]
  const float* Wlin = (const float*)d_in[8];  // [4][128][128]
  float* out = (float*)d_out;

  float* Pt = (float*)d_ws;                   // 128*65536 f32 = 32 MiB
  float* out1_ws = Pt + (size_t)128 * 65536;  // 128*16*256 f32 = 2 MiB

  const int ldsA = 128 * 128 * 4;                    // 64 KiB
  const int ldsB = (65536 + 4096 + 4096 + 256) * 4;  // ~289 KiB
  (void)hipFuncSetAttribute((const void*)pb_u3w3_gemm,
                            hipFuncAttributeMaxDynamicSharedMemorySize, ldsA);
  (void)hipFuncSetAttribute((const void*)pb_channel_fused,
                            hipFuncAttributeMaxDynamicSharedMemorySize, ldsB);

  pb_u3w3_gemm<<<512, 256, ldsA, stream>>>(U3, w3, Pt);
  pb_channel_fused<<<128, 256, ldsB, stream>>>(nf, U1, U2, w1, w2, Pt, out1_ws);
  pb_o3_linear<<<256, 256, 0, stream>>>(out1_ws, Wlin, sc, out);
}